// BertTempRel_54090818126365
// MI455X (gfx1250) — compile-verified
//
#include <hip/hip_runtime.h>
#include <math.h>

typedef float v2f __attribute__((ext_vector_type(2)));
typedef float v8f __attribute__((ext_vector_type(8)));

#define BB 256
#define SS 512
#define DD 768
#define H1 256
#define H2 64
#define NL 4
#define FDIM (4 * DD)   // 3072
#define KT   256        // K tile staged in LDS for GEMM1
#define LDSP 260        // padded LDS row stride (floats): 260 % 64 = 4 -> conflict-free

// ---------------------------------------------------------------------------
// Kernel 1: per-sample features: F[b, base +   0..767] = x[b,0,:] (CLS)
//           F[b, base + 768..1535] = mean over rows [s,e) of x[b]
// one block per sample, 256 threads, each thread covers 3 of 768 columns.
// ---------------------------------------------------------------------------
__global__ void feats_kernel(const float* __restrict__ x,
                             const int* __restrict__ sArr,
                             const int* __restrict__ eArr,
                             float* __restrict__ F, int baseOff) {
    int b = blockIdx.x;
    int t = threadIdx.x;
    int st = sArr[b];
    int en = eArr[b];
    if (en < st + 1) en = st + 1;           // guard empty spans (matches ref)
    float inv = 1.0f / (float)(en - st);
    const float* xb = x + (size_t)b * SS * DD;
    float* Fb = F + (size_t)b * FDIM + baseOff;
    for (int c = t; c < DD; c += 256) {
        float cls = xb[c];                  // row 0 = CLS
        float sum = 0.0f;
        for (int r = st; r < en; ++r)
            sum += xb[(size_t)r * DD + c];
        Fb[c]      = cls;
        Fb[DD + c] = sum * inv;
    }
}

// ---------------------------------------------------------------------------
// GEMM1: C[256,256] = relu(A[256,3072] @ B[3072,256] + bias)
// One 512-thread workgroup (16 wave32s) per 16-row output strip.
// A panel [16, KT] staged in LDS (coalesced global loads, conflict-free
// ds_load reads); wave w computes the 16x16 tile at columns w*16 via
// V_WMMA_F32_16X16X4_F32, streaming its private B columns from global.
// ---------------------------------------------------------------------------
__global__ void gemm1_wmma_lds(const float* __restrict__ A,
                               const float* __restrict__ Bm,
                               const float* __restrict__ bias,
                               float* __restrict__ C) {
    __shared__ float As[16][LDSP];

    const int tid  = threadIdx.x;    // 0..511
    const int wave = tid >> 5;       // 0..15 -> N tile
    const int lane = tid & 31;       // wave32 lane
    const int half = lane >> 4;      // K pair select
    const int lr   = lane & 15;

    const int tm = blockIdx.x * 16;  // output row strip
    const int tn = wave * 16;        // output col tile

    const float* Bcol = Bm + (size_t)(2 * half) * H1 + (tn + lr); // B[2h, tn+lr]

    v8f acc = {};
    for (int k0 = 0; k0 < FDIM; k0 += KT) {
        // ---- stage A[tm..tm+16, k0..k0+KT) into LDS, coalesced ----
#pragma unroll
        for (int i = tid; i < 16 * KT; i += 512) {
            int r = i >> 8;          // i / KT
            int c = i & (KT - 1);    // i % KT
            As[r][c] = A[(size_t)(tm + r) * FDIM + (k0 + c)];
        }
        __syncthreads();

        // ---- 64 WMMAs over this K tile ----
#pragma unroll 8
        for (int kk = 0; kk < KT; kk += 4) {
            v2f a, b;
            a.x = As[lr][kk + 2 * half];
            a.y = As[lr][kk + 2 * half + 1];
            const float* bp = Bcol + (size_t)(k0 + kk) * H1;
            b.x = bp[0];
            b.y = bp[H1];
            acc = __builtin_amdgcn_wmma_f32_16x16x4_f32(
                false, a, false, b, (short)0, acc, false, false);
        }
        __syncthreads();
    }

    // ---- epilogue: bias + relu, C/D layout: VGPR j -> M = j + 8*half ----
    const int n = tn + lr;
    const float bv = bias[n];
#pragma unroll
    for (int j = 0; j < 8; ++j) {
        int m = tm + j + 8 * half;
        C[(size_t)m * H1 + n] = fmaxf(acc[j] + bv, 0.0f);
    }
}

// ---------------------------------------------------------------------------
// Generic small WMMA fp32 GEMM (used for GEMM2): one wave per 16x16 tile.
// ---------------------------------------------------------------------------
__global__ void gemm16_wmma_f32(const float* __restrict__ A,
                                const float* __restrict__ Bm,
                                const float* __restrict__ bias,
                                float* __restrict__ C,
                                int K, int N, int doRelu) {
    const int tm   = blockIdx.x * 16;
    const int tn   = blockIdx.y * 16;
    const int lane = threadIdx.x;        // 0..31 (wave32)
    const int half = lane >> 4;
    const int lr   = lane & 15;

    const float* Arow = A  + (size_t)(tm + lr) * K + 2 * half;
    const float* Bcol = Bm + (size_t)(2 * half) * N + (tn + lr);

    v8f acc = {};
#pragma unroll 8
    for (int k = 0; k < K; k += 4) {
        v2f a, b;
        a.x = Arow[k];
        a.y = Arow[k + 1];
        b.x = Bcol[(size_t)k * N];
        b.y = Bcol[(size_t)(k + 1) * N];
        acc = __builtin_amdgcn_wmma_f32_16x16x4_f32(
            false, a, false, b, (short)0, acc, false, false);
    }

    const int n = tn + lr;
    const float bv = bias[n];
#pragma unroll
    for (int j = 0; j < 8; ++j) {
        int m = tm + j + 8 * half;
        float v = acc[j] + bv;
        if (doRelu) v = fmaxf(v, 0.0f);
        C[(size_t)m * N + n] = v;
    }
}

// ---------------------------------------------------------------------------
// Kernel 3: logits = h2[256,64] @ W3[64,4] + b3 ; softmax over the 4 labels.
// ---------------------------------------------------------------------------
__global__ void head_kernel(const float* __restrict__ h2,
                            const float* __restrict__ W3,
                            const float* __restrict__ b3,
                            float* __restrict__ out) {
    int b = blockIdx.x * blockDim.x + threadIdx.x;
    if (b >= BB) return;
    const float* hb = h2 + (size_t)b * H2;
    float lg[NL];
#pragma unroll
    for (int j = 0; j < NL; ++j) lg[j] = b3[j];
    for (int k = 0; k < H2; ++k) {
        float h = hb[k];
#pragma unroll
        for (int j = 0; j < NL; ++j) lg[j] += h * W3[k * NL + j];
    }
    float mx = fmaxf(fmaxf(lg[0], lg[1]), fmaxf(lg[2], lg[3]));
    float s = 0.0f;
#pragma unroll
    for (int j = 0; j < NL; ++j) { lg[j] = __expf(lg[j] - mx); s += lg[j]; }
    float inv = 1.0f / s;
#pragma unroll
    for (int j = 0; j < NL; ++j) out[(size_t)b * NL + j] = lg[j] * inv;
}

// ---------------------------------------------------------------------------
extern "C" void kernel_launch(void* const* d_in, const int* in_sizes, int n_in,
                              void* d_out, int out_size, void* d_ws, size_t ws_size,
                              hipStream_t stream) {
    const float* x1 = (const float*)d_in[0];
    const float* x2 = (const float*)d_in[1];
    const int*   s1 = (const int*)d_in[2];
    const int*   e1 = (const int*)d_in[3];
    const int*   s2 = (const int*)d_in[4];
    const int*   e2 = (const int*)d_in[5];
    const float* W1 = (const float*)d_in[6];
    const float* b1 = (const float*)d_in[7];
    const float* W2 = (const float*)d_in[8];
    const float* b2 = (const float*)d_in[9];
    const float* W3 = (const float*)d_in[10];
    const float* b3 = (const float*)d_in[11];
    float* out = (float*)d_out;

    // workspace layout
    float* F  = (float*)d_ws;                         // [256, 3072]
    float* h1 = F + (size_t)BB * FDIM;                // [256, 256]
    float* h2 = h1 + (size_t)BB * H1;                 // [256, 64]

    // features: x1 -> cols [0,1536), x2 -> cols [1536,3072)
    feats_kernel<<<BB, 256, 0, stream>>>(x1, s1, e1, F, 0);
    feats_kernel<<<BB, 256, 0, stream>>>(x2, s2, e2, F, 2 * DD);

    // GEMM1: [256,3072]x[3072,256] + b1, relu  (WMMA f32, LDS-staged A panel)
    gemm1_wmma_lds<<<BB / 16, 512, 0, stream>>>(F, W1, b1, h1);

    // GEMM2: [256,256]x[256,64] + b2, relu  (WMMA f32)
    gemm16_wmma_f32<<<dim3(BB / 16, H2 / 16), 32, 0, stream>>>(
        h1, W2, b2, h2, H1, H2, 1);

    // head + softmax
    head_kernel<<<BB / 32, 32, 0, stream>>>(h2, W3, b3, out);
}